// SingleDeformConv2d_77824807403837
// MI455X (gfx1250) — compile-verified
//
#include <hip/hip_runtime.h>

typedef __attribute__((ext_vector_type(2))) float v2f;
typedef __attribute__((ext_vector_type(8))) float v8f;

#define B_   8
#define C_   128
#define H_   64
#define W_   64
#define O_   128
#define KD_  1152              // C_ * 9
#define HW_  4096              // H_ * W_
#define LDA_ 1156              // padded LDS row stride (floats): 1156 % 64 == 4
                               // -> wave ds_load_b64 A-fragment hits all 64 banks once
#define NOFF 32                // offset channels 18, padded to 2 N-tiles
#define TM   32                // positions per block (2 M-subtiles of 16)
#define NPK  (TM * 9)          // bilinear parameter entries (288)

// ---------------------------------------------------------------------------
// Weight reorder (one tiny kernel, both GEMMs):
//   wt   [k = kk*128+c][o]  : main weights   (O,C,3,3) -> 1152x128
//   wtoff[k = kk*128+c][j]  : offset weights (18,C,3,3) -> 1152x32 (zero pad j>=18)
// ---------------------------------------------------------------------------
__global__ __launch_bounds__(256) void reorder_kernel(
    const float* __restrict__ w, const float* __restrict__ w_off,
    float* __restrict__ wt, float* __restrict__ wtoff) {
  int t = blockIdx.x * blockDim.x + threadIdx.x;
  if (t < KD_ * O_) {
    int o = t & 127, k = t >> 7;
    int kk = k >> 7, c = k & 127;
    wt[t] = w[o * KD_ + c * 9 + kk];
  }
  int t2 = t - KD_ * O_;
  if (t2 >= 0 && t2 < KD_ * NOFF) {
    int j = t2 & 31, k = t2 >> 5;
    int kk = k >> 7, c = k & 127;
    wtoff[t2] = (j < 18) ? w_off[j * KD_ + c * 9 + kk] : 0.0f;
  }
}

// ---------------------------------------------------------------------------
// Fused deformable conv: one block = 32 consecutive positions (same b, same h).
//  A) im2col tile P[32][1152] -> LDS
//  B) waves 0-3: offset GEMM (32x32) on WMMA, offsets stay in LDS
//  C) bilinear coefficients from LDS offsets
//  D) overwrite LDS tile with deformably sampled A
//  E) 8 waves: main GEMM; each wave = one 16-wide N-slice x 2 M-subtiles,
//     so every B-fragment load feeds two v_wmma_f32_16x16x4_f32 (halves the
//     L2 weight traffic that would otherwise dominate the roofline).
// ---------------------------------------------------------------------------
__global__ __launch_bounds__(256) void deform_fused_kernel(
    const float* __restrict__ x, const float* __restrict__ b_off,
    const float* __restrict__ wtoff, const float* __restrict__ wt,
    float* __restrict__ out) {
  extern __shared__ float smem[];
  float* sA   = smem;                         // [TM][LDA_] im2col / sampled tile
  float* sOff = smem + TM * LDA_;             // [TM][NOFF] offset-conv results
  int*   sY0  = (int*)(sOff + TM * NOFF);     // bilinear params, NPK entries each
  int*   sY1  = sY0 + NPK;
  int*   sX0  = sY1 + NPK;
  int*   sX1  = sX0 + NPK;
  float* sW00 = (float*)(sX1 + NPK);
  float* sW01 = sW00 + NPK;
  float* sW10 = sW01 + NPK;
  float* sW11 = sW10 + NPK;

  const int tid = threadIdx.x;
  const int tileBase = blockIdx.x * TM;       // flat position b*4096 + h*64 + w0
  const int b0 = tileBase >> 12;
  const int hw0 = tileBase & 4095;
  const int h0 = hw0 >> 6, w0 = hw0 & 63;     // all TM positions share b0,h0
  const float* xb = x + (size_t)b0 * C_ * HW_;

  // ---- Phase A: plain im2col tile for the offset conv --------------------
  // i = (c*9 + kk)*TM + m : consecutive lanes -> consecutive m (coalesced x).
  for (int i = tid; i < TM * KD_; i += 256) {
    int m  = i & (TM - 1);
    int r  = i >> 5;                          // c*9 + kk
    int kk = r % 9;
    int c  = r / 9;
    int yy = h0 - 1 + kk / 3;
    int xx = w0 + m - 1 + kk % 3;
    bool v = (yy >= 0) && (yy < H_) && (xx >= 0) && (xx < W_);
    sA[m * LDA_ + kk * 128 + c] = v ? xb[(c << 12) + (yy << 6) + xx] : 0.0f;
  }
  __syncthreads();

  const int wave  = tid >> 5;
  const int lane  = tid & 31;
  const int nloc  = lane & 15;                // N within tile; A-row M = lane%16
  const int khalf = (lane >> 4) << 1;         // lanes 0-15: K{0,1}; 16-31: K{2,3}

  // ---- Phase B: offset GEMM on WMMA (waves 0-3: 2 M-subtiles x 2 N-tiles) -
  if (wave < 4) {
    int mt = wave >> 1;                       // M-subtile 0/1
    int j  = ((wave & 1) << 4) + nloc;        // offset channel (valid < 18)
    const float* arow = sA + (mt * 16 + nloc) * LDA_;
    v8f acc = {};
#pragma unroll 4
    for (int k = 0; k < KD_; k += 4) {
      v2f a = *(const v2f*)&arow[k + khalf];
      const float* wp = wtoff + (k + khalf) * NOFF + j;
      v2f bf;
      bf.x = wp[0];
      bf.y = wp[NOFF];
      acc = __builtin_amdgcn_wmma_f32_16x16x4_f32(
          false, a, false, bf, (short)0, acc, false, false);
    }
#pragma unroll
    for (int r = 0; r < 8; ++r) {
      int m = mt * 16 + r + ((lane >> 4) << 3);
      sOff[m * NOFF + j] = acc[r];
    }
  }
  __syncthreads();

  // ---- Phase C: bilinear coefficients from LDS offsets -------------------
  if (tid < NPK) {                            // NPK = 288 > 256: handle tail below
    // handled in loop form to cover all 288 entries with 256 threads
  }
  for (int pkm = tid; pkm < NPK; pkm += 256) {
    int m = pkm / 9, kk = pkm % 9;
    float oy = sOff[m * NOFF + 2 * kk]     + b_off[2 * kk];
    float ox = sOff[m * NOFF + 2 * kk + 1] + b_off[2 * kk + 1];
    float py = (float)(h0 - 1 + kk / 3) + oy;
    float px = (float)(w0 + m - 1 + kk % 3) + ox;
    float y0f = floorf(py), x0f = floorf(px);
    float dy = py - y0f, dx = px - x0f;
    int y0 = (int)y0f, x0 = (int)x0f;
    int y1 = y0 + 1,   x1 = x0 + 1;
    float vy0 = (y0 >= 0 && y0 < H_) ? 1.0f : 0.0f;
    float vy1 = (y1 >= 0 && y1 < H_) ? 1.0f : 0.0f;
    float vx0 = (x0 >= 0 && x0 < W_) ? 1.0f : 0.0f;
    float vx1 = (x1 >= 0 && x1 < W_) ? 1.0f : 0.0f;
    sW00[pkm] = (1.0f - dy) * (1.0f - dx) * vy0 * vx0;
    sW01[pkm] = (1.0f - dy) * dx          * vy0 * vx1;
    sW10[pkm] = dy * (1.0f - dx)          * vy1 * vx0;
    sW11[pkm] = dy * dx                   * vy1 * vx1;
    sY0[pkm] = min(max(y0, 0), H_ - 1);
    sY1[pkm] = min(max(y1, 0), H_ - 1);
    sX0[pkm] = min(max(x0, 0), W_ - 1);
    sX1[pkm] = min(max(x1, 0), W_ - 1);
  }
  __syncthreads();

  // ---- Phase D: deformable bilinear sampling, overwrite sA ---------------
  for (int i = tid; i < TM * KD_; i += 256) {
    int m  = i & (TM - 1);
    int r  = i >> 5;
    int kk = r % 9;
    int c  = r / 9;
    int pk = m * 9 + kk;
    const float* xc = xb + (c << 12);
    int y0 = sY0[pk], y1 = sY1[pk], xa = sX0[pk], xb1 = sX1[pk];
    float v = sW00[pk] * xc[(y0 << 6) + xa] + sW01[pk] * xc[(y0 << 6) + xb1]
            + sW10[pk] * xc[(y1 << 6) + xa] + sW11[pk] * xc[(y1 << 6) + xb1];
    sA[m * LDA_ + kk * 128 + c] = v;
  }
  __syncthreads();

  // ---- Phase E: main GEMM; 1 N-slice x 2 M-subtiles per wave -------------
  const int n = (wave << 4) + nloc;           // output channel
  const float* arow0 = sA + nloc * LDA_;
  const float* arow1 = arow0 + 16 * LDA_;
  v8f acc0 = {};
  v8f acc1 = {};
#pragma unroll 2
  for (int k = 0; k < KD_; k += 4) {
    const float* wp = wt + (k + khalf) * O_ + n;
    v2f bf;
    bf.x = wp[0];
    bf.y = wp[O_];
    v2f a0 = *(const v2f*)&arow0[k + khalf];  // ds_load_b64, bank-conflict-free
    v2f a1 = *(const v2f*)&arow1[k + khalf];
    acc0 = __builtin_amdgcn_wmma_f32_16x16x4_f32(
        false, a0, false, bf, (short)0, acc0, false, false);
    acc1 = __builtin_amdgcn_wmma_f32_16x16x4_f32(
        false, a1, false, bf, (short)0, acc1, false, false);
  }

  // D layout: VGPR r -> (M = r + 8*(lane/16), N = lane%16); per-lane outputs
  // are 8 consecutive floats of one out row -> compiler merges to b128 stores.
#pragma unroll
  for (int r = 0; r < 8; ++r) {
    int m   = r + ((lane >> 4) << 3);
    int pos = tileBase + m;
    int bb  = pos >> 12;
    int phw = pos & 4095;
    out[((bb * O_ + n) << 12) + phw] = acc0[r];
  }
#pragma unroll
  for (int r = 0; r < 8; ++r) {
    int m   = 16 + r + ((lane >> 4) << 3);
    int pos = tileBase + m;
    int bb  = pos >> 12;
    int phw = pos & 4095;
    out[((bb * O_ + n) << 12) + phw] = acc1[r];
  }
}

// ---------------------------------------------------------------------------
extern "C" void kernel_launch(void* const* d_in, const int* in_sizes, int n_in,
                              void* d_out, int out_size, void* d_ws, size_t ws_size,
                              hipStream_t stream) {
  const float* x     = (const float*)d_in[0];
  const float* w_off = (const float*)d_in[1];
  const float* b_off = (const float*)d_in[2];
  const float* w     = (const float*)d_in[3];
  float* out = (float*)d_out;

  // Workspace: wt (1152x128) then wtoff (1152x32)  (~0.72 MB total)
  float* ws_wt    = (float*)d_ws;
  float* ws_wtoff = ws_wt + (size_t)KD_ * O_;

  int reorder_items = KD_ * O_ + KD_ * NOFF;
  reorder_kernel<<<(reorder_items + 255) / 256, 256, 0, stream>>>(
      w, w_off, ws_wt, ws_wtoff);

  size_t shmem = (size_t)TM * LDA_ * sizeof(float)   // sA       (147,968 B)
               + (size_t)TM * NOFF * sizeof(float)   // sOff     (  4,096 B)
               + (size_t)NPK * 8 * sizeof(float);    // params   (  9,216 B)
  deform_fused_kernel<<<(B_ * HW_) / TM, 256, shmem, stream>>>(
      x, b_off, ws_wtoff, ws_wt, out);
}